// DistanceAwareInterpolator_23871428231187
// MI455X (gfx1250) — compile-verified
//
#include <hip/hip_runtime.h>
#include <stdint.h>

// ---------------------------------------------------------------------------
// DistanceAwareInterpolator for MI455X (gfx1250).
// Pure bandwidth problem: ~150 MB HBM traffic -> ~6.5us floor @ 23.3 TB/s.
// Per-bin weights/indices are batch-invariant -> precompute once (kernel 1,
// 64KB table, L2-resident), then stream (kernel 2): async-to-LDS staging of
// each 4KB H row, ds_load_b64 gathers, nontemporal b128 output stores.
// WMMA deliberately NOT used: banded-sparse matmul formulation would do
// ~250x excess FLOPs and still be HBM-bound.
// ---------------------------------------------------------------------------

#if __has_builtin(__builtin_amdgcn_global_load_async_to_lds_b128) && \
    __has_builtin(__builtin_amdgcn_s_wait_asynccnt)
#define USE_ASYNC_LDS 1
#else
#define USE_ASYNC_LDS 0
#endif

// Native clang vector types (required by nontemporal/async builtins; the
// HIP_vector_type wrapper classes are rejected).
typedef float nfloat4 __attribute__((ext_vector_type(4)));
typedef float nfloat2 __attribute__((ext_vector_type(2)));
typedef int   nint4   __attribute__((ext_vector_type(4)));

#if USE_ASYNC_LDS
typedef __attribute__((address_space(1))) nint4 g_nint4;  // global b128 src
typedef __attribute__((address_space(3))) nint4 l_nint4;  // LDS b128 dst
#endif

__device__ __forceinline__ float softplus_ref(float x) {
  // jax.nn.softplus(x) = max(x,0) + log1p(exp(-|x|))
  return fmaxf(x, 0.0f) + log1pf(expf(-fabsf(x)));
}

// upper_bound (searchsorted side='right') over locext[0..n-1]
__device__ __forceinline__ int upper_bound_lds(const float* locext, int n, float v) {
  int lo = 0, hi = n;
  while (lo < hi) {
    int mid = (lo + hi) >> 1;
    if (locext[mid] <= v) lo = mid + 1; else hi = mid;
  }
  return lo;
}

// Compute per-bin coefficients: a0,a1 = normalized weights; o0,o1 = LDS byte
// offsets of the two bracketing float2 entries in the staged H-row layout:
//   slot p in [0,NP)   : pilot p          (byte offset p*8)
//   slot NP            : h0 extrapolation (byte offset NP*8)
//   slot NP+1          : hN extrapolation (byte offset (NP+1)*8)
__device__ __forceinline__ void coef_from_search(int i, const float* locext, int NP,
                                                 float decay, float& a0, float& a1,
                                                 int& o0, int& o1) {
  float fi = (float)i;
  int lo = upper_bound_lds(locext, NP + 2, fi);
  int left = lo - 1;
  left = left < 0 ? 0 : (left > NP ? NP : left);
  int right = left + 1;
  float x0 = locext[left], x1 = locext[right];
  float wl = expf(-decay * fabsf(fi - x0));
  float wr = expf(-decay * fabsf(x1 - fi));
  float w = wl + wr + 1e-12f;
  a0 = wl / w;
  a1 = wr / w;
  int slot0 = (left == 0) ? NP : (left - 1);
  int slot1 = (right == NP + 1) ? (NP + 1) : (right - 1);
  o0 = slot0 * 8;
  o1 = slot1 * 8;
}

// ---------------------------------------------------------------------------
// Kernel 1: per-bin coefficient table (Nfft entries x 16B) into d_ws.
// ---------------------------------------------------------------------------
__global__ void dai_coeff_kernel(const float* __restrict__ pilot_pos,
                                 const float* __restrict__ decay_param,
                                 nfloat4* __restrict__ coef,
                                 int NP, int Nfft) {
  extern __shared__ float locext[];  // NP+2 floats
  const int tid = threadIdx.x;
  for (int p = tid; p < NP; p += blockDim.x) locext[1 + p] = pilot_pos[p] - 1.0f;
  if (tid == 0) {
    locext[0] = 0.0f;
    locext[NP + 1] = (float)(Nfft - 1);
  }
  __syncthreads();

  int i = blockIdx.x * blockDim.x + tid;
  if (i >= Nfft) return;

  float decay = softplus_ref(decay_param[0]);
  float a0, a1;
  int o0, o1;
  coef_from_search(i, locext, NP, decay, a0, a1, o0, o1);

  nfloat4 r;
  r.x = a0;
  r.y = a1;
  r.z = __int_as_float(o0);
  r.w = __int_as_float(o1);
  coef[i] = r;
}

// ---------------------------------------------------------------------------
// Kernel 2: one block per batch row b. Stage H row (NP*2 floats) into LDS
// via async b128 copies, append h0/hN slots, then stream Nfft*2 outputs.
// ---------------------------------------------------------------------------
template <bool PRECOMP>
__global__ void dai_interp_kernel(const float* __restrict__ LS,
                                  const float* __restrict__ pilot_pos,
                                  const float* __restrict__ decay_param,
                                  const nfloat4* __restrict__ coef,
                                  float* __restrict__ out,
                                  int NP, int Nfft) {
  extern __shared__ float smem[];
  float* hx = smem;                          // 2*(NP+2) floats
  float* locext = smem + 2 * (NP + 2);       // NP+2 floats (only if !PRECOMP)
  const int tid = threadIdx.x;
  const int b = blockIdx.x;
  const float* gsrc = LS + (size_t)b * NP * 2;

  // --- stage H row into LDS slots [0, 2*NP) via async tensor-path copies ---
  const int nchunk = (NP * 2) >> 2;  // b128 chunks
  for (int c = tid; c < nchunk; c += blockDim.x) {
#if USE_ASYNC_LDS
    __builtin_amdgcn_global_load_async_to_lds_b128(
        (g_nint4*)(uintptr_t)(gsrc + c * 4),
        (l_nint4*)(uint32_t)(uintptr_t)(hx + c * 4),
        /*offset=*/0, /*cpol=*/0);
#else
    *(nfloat4*)(hx + c * 4) = *(const nfloat4*)(gsrc + c * 4);
#endif
  }
  for (int e = nchunk * 4 + tid; e < NP * 2; e += blockDim.x) hx[e] = gsrc[e];

  if (!PRECOMP) {
    for (int p = tid; p < NP; p += blockDim.x) locext[1 + p] = pilot_pos[p] - 1.0f;
  }

  // Thread 0 computes the two extrapolated endpoints straight from global
  // (overlaps with the in-flight async copy; independent of LDS contents).
  if (tid == 0) {
    if (!PRECOMP) {
      locext[0] = 0.0f;
      locext[NP + 1] = (float)(Nfft - 1);
    }
    float l0 = pilot_pos[0] - 1.0f, l1 = pilot_pos[1] - 1.0f;
    float lm2 = pilot_pos[NP - 2] - 1.0f, lm1 = pilot_pos[NP - 1] - 1.0f;
    float H00 = gsrc[0], H01 = gsrc[1];
    float H10 = gsrc[2], H11 = gsrc[3];
    float Hm20 = gsrc[(NP - 2) * 2], Hm21 = gsrc[(NP - 2) * 2 + 1];
    float Hm10 = gsrc[(NP - 1) * 2], Hm11 = gsrc[(NP - 1) * 2 + 1];
    float inv_dl = 1.0f / (l1 - l0);
    float inv_dr = 1.0f / (lm1 - lm2);
    // h0 = H[:,0] - slope_l * loc[0]
    hx[2 * NP + 0] = H00 - (H10 - H00) * inv_dl * l0;
    hx[2 * NP + 1] = H01 - (H11 - H01) * inv_dl * l0;
    // hN = H[:,-1] + slope_r * (Nfft-1 - loc[-1])
    float dr = (float)(Nfft - 1) - lm1;
    hx[2 * NP + 2] = Hm10 + (Hm10 - Hm20) * inv_dr * dr;
    hx[2 * NP + 3] = Hm11 + (Hm11 - Hm21) * inv_dr * dr;
  }

#if USE_ASYNC_LDS
  __builtin_amdgcn_s_wait_asynccnt(0);
#endif
  __syncthreads();

  float decay = 0.0f;
  if (!PRECOMP) decay = softplus_ref(decay_param[0]);

  float* outrow = out + (size_t)b * Nfft * 2;
  const int npair = Nfft >> 1;

#pragma unroll 2
  for (int pi = tid; pi < npair; pi += blockDim.x) {
    const int i0 = pi * 2;
    float a0x, a1x, a0y, a1y;
    int o00, o01, o10, o11;
    if (PRECOMP) {
      nfloat4 c0 = coef[i0];
      nfloat4 c1 = coef[i0 + 1];
      a0x = c0.x; a1x = c0.y; o00 = __float_as_int(c0.z); o01 = __float_as_int(c0.w);
      a0y = c1.x; a1y = c1.y; o10 = __float_as_int(c1.z); o11 = __float_as_int(c1.w);
    } else {
      coef_from_search(i0,     locext, NP, decay, a0x, a1x, o00, o01);
      coef_from_search(i0 + 1, locext, NP, decay, a0y, a1y, o10, o11);
    }
    nfloat2 p00 = *(const nfloat2*)((const char*)hx + o00);
    nfloat2 p01 = *(const nfloat2*)((const char*)hx + o01);
    nfloat2 p10 = *(const nfloat2*)((const char*)hx + o10);
    nfloat2 p11 = *(const nfloat2*)((const char*)hx + o11);
    nfloat4 r;
    r.x = a0x * p00.x + a1x * p01.x;
    r.y = a0x * p00.y + a1x * p01.y;
    r.z = a0y * p10.x + a1y * p11.x;
    r.w = a0y * p10.y + a1y * p11.y;
    // 134MB streaming output with zero reuse: bypass caches (non-temporal).
    __builtin_nontemporal_store(r, (nfloat4*)(outrow + i0 * 2));
  }

  // Odd-Nfft tail (not hit for Nfft=4096).
  for (int i = npair * 2 + tid; i < Nfft; i += blockDim.x) {
    float a0, a1;
    int o0, o1;
    if (PRECOMP) {
      nfloat4 c0 = coef[i];
      a0 = c0.x; a1 = c0.y; o0 = __float_as_int(c0.z); o1 = __float_as_int(c0.w);
    } else {
      coef_from_search(i, locext, NP, decay, a0, a1, o0, o1);
    }
    nfloat2 p0 = *(const nfloat2*)((const char*)hx + o0);
    nfloat2 p1 = *(const nfloat2*)((const char*)hx + o1);
    nfloat2 r;
    r.x = a0 * p0.x + a1 * p1.x;
    r.y = a0 * p0.y + a1 * p1.y;
    __builtin_nontemporal_store(r, (nfloat2*)(outrow + i * 2));
  }
}

// ---------------------------------------------------------------------------
// Host launcher
// ---------------------------------------------------------------------------
extern "C" void kernel_launch(void* const* d_in, const int* in_sizes, int n_in,
                              void* d_out, int out_size, void* d_ws, size_t ws_size,
                              hipStream_t stream) {
  const float* LS          = (const float*)d_in[0];  // (B, NP, 2) f32
  const float* pilot_pos   = (const float*)d_in[1];  // (NP,) f32
  const float* decay_param = (const float*)d_in[2];  // (1,) f32
  // d_in[3] is Nfft as a device int; derive all shapes host-side instead:
  const int NP   = in_sizes[1];
  const int B    = in_sizes[0] / (2 * NP);
  const int Nfft = out_size / (2 * B);
  float* out = (float*)d_out;

  const int threads = 256;
  const size_t coefBytes = (size_t)Nfft * 16;
  const bool precomp = (ws_size >= coefBytes);

  if (precomp) {
    const int blocks1 = (Nfft + threads - 1) / threads;
    const size_t lds1 = (size_t)(NP + 2) * sizeof(float);
    dai_coeff_kernel<<<blocks1, threads, lds1, stream>>>(
        pilot_pos, decay_param, (nfloat4*)d_ws, NP, Nfft);

    const size_t lds2 = (size_t)(2 * (NP + 2)) * sizeof(float);
    dai_interp_kernel<true><<<B, threads, lds2, stream>>>(
        LS, pilot_pos, decay_param, (const nfloat4*)d_ws, out, NP, Nfft);
  } else {
    const size_t lds2 = (size_t)(3 * (NP + 2)) * sizeof(float);
    dai_interp_kernel<false><<<B, threads, lds2, stream>>>(
        LS, pilot_pos, decay_param, nullptr, out, NP, Nfft);
  }
}